// RetinaFaceLoss_7017976562193
// MI455X (gfx1250) — compile-verified
//
#include <hip/hip_runtime.h>

typedef __attribute__((ext_vector_type(2))) float v2f;
typedef __attribute__((ext_vector_type(8))) float v8f;

#define THREADS 256
#define NBLK 64      // x-blocks per batch; multiple of 4 for the WMMA reduction
#define MAXG 32

__device__ __forceinline__ float smooth_l1(float d) {
    float a = fabsf(d);
    return a < 1.0f ? 0.5f * d * d : a - 0.5f;
}

// ---------------------------------------------------------------------------
// Main kernel: grid = (NBLK, B). Each block streams a strided slice of the
// A=120000 anchors for one batch. GT boxes/landmarks live in LDS (broadcast
// reads). Negatives (the overwhelming majority) never touch pred_bbox /
// pred_landmarks, collapsing HBM traffic from ~125MB to ~17MB + L2-resident
// anchors. Block partials go to ws[4][B][NBLK].
// ---------------------------------------------------------------------------
__global__ __launch_bounds__(THREADS) void rf_loss_main(
    const float* __restrict__ pred_cls,   // [B,A,2]
    const float* __restrict__ pred_bbox,  // [B,A,4]
    const float* __restrict__ pred_ldm,   // [B,A,10]
    const float* __restrict__ anchors,    // [A,4]
    const float* __restrict__ gt_boxes,   // [B,G,4]
    const float* __restrict__ gt_ldm,     // [B,G,10]
    float* __restrict__ ws,               // [4][B][NBLK]
    int A, int B, int G)
{
    const int b   = blockIdx.y;
    const int tid = threadIdx.x;

    __shared__ float4 s_gtb[MAXG];        // gt boxes
    __shared__ float  s_area[MAXG];       // gt areas
    __shared__ float  s_gl[MAXG * 10];    // gt landmarks
    __shared__ float4 s_red[THREADS];     // block reduction

    if (tid < G)
        s_gtb[tid] = ((const float4*)gt_boxes)[(size_t)b * G + tid];
    for (int i = tid; i < G * 10; i += THREADS)
        s_gl[i] = gt_ldm[(size_t)b * G * 10 + i];
    __syncthreads();
    if (tid < G) {
        float4 g = s_gtb[tid];
        s_area[tid] = (g.z - g.x) * (g.w - g.y);
    }
    __syncthreads();

    float ce_acc = 0.0f, np_acc = 0.0f, rg_acc = 0.0f, ld_acc = 0.0f;
    const int stride = gridDim.x * THREADS;

    for (int a = blockIdx.x * THREADS + tid; a < A; a += stride) {
        // prefetch next iteration's streaming data (global_prefetch_b8)
        int an = a + stride;
        if (an < A) {
            __builtin_prefetch(anchors + (size_t)an * 4, 0, 1);
            __builtin_prefetch(pred_cls + ((size_t)b * A + an) * 2, 0, 1);
        }

        float4 anc = ((const float4*)anchors)[a];          // global_load_b128
        float aw = anc.z - anc.x, ah = anc.w - anc.y;
        float area_a = aw * ah;

        float best = -1.0f;
        int   bi   = 0;
        #pragma unroll 4
        for (int g = 0; g < G; ++g) {
            float4 gb = s_gtb[g];                           // LDS broadcast
            float iw = fminf(anc.z, gb.z) - fmaxf(anc.x, gb.x);
            float ih = fminf(anc.w, gb.w) - fmaxf(anc.y, gb.y);
            iw = fmaxf(iw, 0.0f);
            ih = fmaxf(ih, 0.0f);
            float inter = iw * ih;
            float iou = inter / (area_a + s_area[g] - inter);
            if (iou > best) { best = iou; bi = g; }         // first-max like jnp.argmax
        }
        bool pos = best >= 0.5f;

        // cross-entropy over 2 classes: ce = lse - logit[target]
        float2 c = *(const float2*)(pred_cls + ((size_t)b * A + a) * 2);
        float mx  = fmaxf(c.x, c.y);
        float lse = mx + logf(expf(c.x - mx) + expf(c.y - mx));
        ce_acc += lse - (pos ? c.y : c.x);

        if (pos) {                                          // rare path
            np_acc += 1.0f;
            float4 gb = s_gtb[bi];
            float gw = gb.z - gb.x, gh = gb.w - gb.y;
            float gcx = gb.x + 0.5f * gw, gcy = gb.y + 0.5f * gh;
            float acx = anc.x + 0.5f * aw, acy = anc.y + 0.5f * ah;
            float t0 = (gcx - acx) / aw;
            float t1 = (gcy - acy) / ah;
            float t2 = logf(gw / aw);
            float t3 = logf(gh / ah);
            float4 pb = *(const float4*)(pred_bbox + ((size_t)b * A + a) * 4);
            rg_acc += smooth_l1(pb.x - t0) + smooth_l1(pb.y - t1) +
                      smooth_l1(pb.z - t2) + smooth_l1(pb.w - t3);
            size_t li = ((size_t)b * A + a) * 10;           // 8B aligned
            #pragma unroll
            for (int j = 0; j < 5; ++j) {
                float2 p = *(const float2*)(pred_ldm + li + j * 2);
                ld_acc += smooth_l1(p.x - s_gl[bi * 10 + j * 2]) +
                          smooth_l1(p.y - s_gl[bi * 10 + j * 2 + 1]);
            }
        }
    }

    s_red[tid] = make_float4(ce_acc, np_acc, rg_acc, ld_acc);
    __syncthreads();
    for (int s = THREADS / 2; s > 0; s >>= 1) {
        if (tid < s) {
            float4 o = s_red[tid + s];
            float4 m = s_red[tid];
            m.x += o.x; m.y += o.y; m.z += o.z; m.w += o.w;
            s_red[tid] = m;
        }
        __syncthreads();
    }
    if (tid == 0) {
        float4 r = s_red[0];
        ws[(0 * B + b) * NBLK + blockIdx.x] = r.x;   // ce sum
        ws[(1 * B + b) * NBLK + blockIdx.x] = r.y;   // npos
        ws[(2 * B + b) * NBLK + blockIdx.x] = r.z;   // reg sum
        ws[(3 * B + b) * NBLK + blockIdx.x] = r.w;   // ldm sum
    }
}

// ---------------------------------------------------------------------------
// Finalize: one wave32 (EXEC all-ones as WMMA requires). Cross-block
// reduction via V_WMMA_F32_16X16X4_F32: A = ones(16x4) makes every D row the
// column sums of B. B-operand column n = lane&15, so column n accumulates
// exactly batch-n partials (fed from lanes n and n+16), 4 blocks per WMMA.
// ---------------------------------------------------------------------------
__global__ __launch_bounds__(32) void rf_loss_finalize(
    const float* __restrict__ ws, float* __restrict__ out, int A, int B)
{
    const int lane = threadIdx.x;
    __shared__ float s_sum[4][16];

    v2f ones; ones.x = 1.0f; ones.y = 1.0f;
    const int n  = min(lane & 15, B - 1);   // batch column (clamped for safety)
    const int hi = lane >> 4;               // 0: K=0,1 rows, 1: K=2,3 rows

    for (int q = 0; q < 4; ++q) {
        const float* base = ws + ((size_t)q * B + n) * NBLK;
        v8f acc = {};
        for (int t = 0; t < NBLK; t += 4) {
            v2f bv;
            bv.x = base[t + hi * 2 + 0];
            bv.y = base[t + hi * 2 + 1];
            // D = ones(16x4) * B(4x16) + C  ->  every row = per-batch sums
            acc = __builtin_amdgcn_wmma_f32_16x16x4_f32(
                false, ones, false, bv, (short)0, acc, false, false);
        }
        if (lane < 16) s_sum[q][lane] = acc[0];
        __syncthreads();
    }

    if (lane == 0) {
        float cls = 0.0f, reg = 0.0f, ldm = 0.0f;
        for (int b2 = 0; b2 < B; ++b2) {
            cls += s_sum[0][b2] / (float)A;
            float npos  = s_sum[1][b2];
            bool  has   = npos > 0.0f;
            float denom = fmaxf(npos, 1.0f);
            reg += has ? s_sum[2][b2] / (denom * 4.0f)  : 0.0f;
            ldm += has ? s_sum[3][b2] / (denom * 10.0f) : 0.0f;
        }
        cls /= (float)B; reg /= (float)B; ldm /= (float)B;
        out[0] = cls + reg + ldm;
        out[1] = cls;
        out[2] = reg;
        out[3] = ldm;
    }
}

extern "C" void kernel_launch(void* const* d_in, const int* in_sizes, int n_in,
                              void* d_out, int out_size, void* d_ws, size_t ws_size,
                              hipStream_t stream) {
    const float* pred_cls  = (const float*)d_in[0];
    const float* pred_bbox = (const float*)d_in[1];
    const float* pred_ldm  = (const float*)d_in[2];
    const float* anchors   = (const float*)d_in[3];
    const float* gt_boxes  = (const float*)d_in[4];
    const float* gt_ldm    = (const float*)d_in[5];

    const int A = in_sizes[3] / 4;          // 120000
    const int B = in_sizes[0] / (A * 2);    // 16
    const int G = in_sizes[4] / (B * 4);    // 32

    float* ws  = (float*)d_ws;              // uses 4*B*NBLK floats = 16 KB
    float* out = (float*)d_out;

    dim3 grid(NBLK, B);
    rf_loss_main<<<grid, THREADS, 0, stream>>>(
        pred_cls, pred_bbox, pred_ldm, anchors, gt_boxes, gt_ldm, ws, A, B, G);
    rf_loss_finalize<<<1, 32, 0, stream>>>(ws, out, A, B);
}